// NNHom_55087250538542
// MI455X (gfx1250) — compile-verified
//
#include <hip/hip_runtime.h>
#include <hip/hip_bf16.h>

// Problem constants (match reference setup_inputs exactly)
#define NB    128                 // batch
#define NSITE 4096                // 64*64 lattice sites
#define DIMG  15                  // dim su(4)
#define NTRIP (NB * NSITE)        // 524288 (b,site) triples

typedef __attribute__((ext_vector_type(2))) float v2f;
typedef __attribute__((ext_vector_type(8))) float v8f;

// ---------------------------------------------------------------------------
// rho(i*T_k) entries: realification of i * (su(4) Gell-Mann-style basis).
// k in [0,6):   T real antisymmetric (E[j,l]=1,E[l,j]=-1); H=iT -> [[0,-T],[T,0]]
// k in [6,12):  T = i*S, S symmetric (S[j,l]=S[l,j]=1);    H=-S -> [[-S,0],[0,-S]]
// k in [12,15): T = i*Dt/sqrt(d(d+1));                     H=-Dt/s -> diag blocks
// ---------------------------------------------------------------------------
__device__ __forceinline__ float rho_entry(int k, int r, int c) {
    const int PJ[6] = {0, 0, 0, 1, 1, 2};
    const int PL[6] = {1, 2, 3, 2, 3, 3};
    if (k < 6) {
        int j = PJ[k], l = PL[k];
        if (r < 4 && c >= 4) {          // top-right = -T
            int cc = c - 4;
            if (r == j && cc == l) return -1.0f;
            if (r == l && cc == j) return 1.0f;
            return 0.0f;
        }
        if (r >= 4 && c < 4) {          // bottom-left = +T
            int rr = r - 4;
            if (rr == j && c == l) return 1.0f;
            if (rr == l && c == j) return -1.0f;
            return 0.0f;
        }
        return 0.0f;
    } else if (k < 12) {
        int j = PJ[k - 6], l = PL[k - 6];
        if ((r < 4) != (c < 4)) return 0.0f;
        int rr = r & 3, cc = c & 3;
        return ((rr == j && cc == l) || (rr == l && cc == j)) ? -1.0f : 0.0f;
    } else {
        if (r != c) return 0.0f;
        int d = k - 11;                 // 1..3
        int rr = r & 3;
        float s = (d == 1) ? 0.70710678118f   // 1/sqrt(2)
                : (d == 2) ? 0.40824829046f   // 1/sqrt(6)
                           : 0.28867513459f;  // 1/sqrt(12)
        float v = (rr < d) ? 1.0f : (rr == d) ? (float)(-d) : 0.0f;
        return -v * s;
    }
}

// ---------------------------------------------------------------------------
// Kernel 1: a_[site][8x8] = sum_k coeffs[site][k] * rho(iT_k)
// GEMM [4096 x 16(pad)] @ [16 x 64] via V_WMMA_F32_16X16X4_F32.
// One wave -> 16 sites x 64 cols (4 n-tiles, 4 k-steps of K=4).
// ---------------------------------------------------------------------------
__global__ __launch_bounds__(128) void build_a_kernel(
    const float* __restrict__ coeffs,   // [4096][15]
    float* __restrict__ a_ws)           // [4096][64]
{
    __shared__ float Bmat[16 * 64];     // rows 0..14 = rho flat, row 15 = 0
    for (int e = threadIdx.x; e < 16 * 64; e += 128) {
        int k = e >> 6, n = e & 63;
        Bmat[e] = (k < DIMG) ? rho_entry(k, n >> 3, n & 7) : 0.0f;
    }
    __syncthreads();

    const int wave  = threadIdx.x >> 5;
    const int lane  = threadIdx.x & 31;
    const int tile  = blockIdx.x * 4 + wave;   // 0..255, 16 sites each
    const int mrow  = lane & 15;               // M (site) / N (col) sub-index
    const int hi    = lane >> 4;               // lane half
    const int khalf = hi << 1;                 // K offset 0 or 2 within a step
    const int site_m = tile * 16 + mrow;

    v8f acc[4];
    #pragma unroll
    for (int nt = 0; nt < 4; ++nt) acc[nt] = (v8f){0,0,0,0,0,0,0,0};

    #pragma unroll
    for (int kk = 0; kk < 4; ++kk) {
        const int k0 = kk * 4 + khalf;         // this lane-half's K pair
        v2f a;
        a.x = (k0 + 0 < DIMG) ? coeffs[site_m * DIMG + k0 + 0] : 0.0f;
        a.y = (k0 + 1 < DIMG) ? coeffs[site_m * DIMG + k0 + 1] : 0.0f;
        #pragma unroll
        for (int nt = 0; nt < 4; ++nt) {
            v2f b;
            b.x = Bmat[(k0 + 0) * 64 + nt * 16 + mrow];
            b.y = Bmat[(k0 + 1) * 64 + nt * 16 + mrow];   // row 15 is zeros
            acc[nt] = __builtin_amdgcn_wmma_f32_16x16x4_f32(
                false, a, false, b, (short)0, acc[nt], false, false);
        }
    }

    // D layout: VGPR r -> M = r (lanes 0-15) / r+8 (lanes 16-31), N = lane&15
    #pragma unroll
    for (int nt = 0; nt < 4; ++nt) {
        #pragma unroll
        for (int r = 0; r < 8; ++r) {
            int site = tile * 16 + r + hi * 8;
            a_ws[site * 64 + nt * 16 + mrow] = acc[nt][r];
        }
    }
}

// ---------------------------------------------------------------------------
// complex helpers
// ---------------------------------------------------------------------------
__device__ __forceinline__ float2 cmul(float2 a, float2 b) {
    return make_float2(a.x * b.x - a.y * b.y, a.x * b.y + a.y * b.x);
}

// ---------------------------------------------------------------------------
// Kernel 2: per (b,site) triple.
//   Y = a_ X ; w = a_ Y ; lam = sqrt(1+YtY) ; tildeZ = lam X + iY
//   J = lam I - X w^T / lam + i a_ ;  det(J) via unrolled no-pivot complex LU
//   d_site = det / lam^2
// J ~ I + O(0.05) perturbation -> diagonally dominant, pivot-free LU is safe.
// ---------------------------------------------------------------------------
__global__ __launch_bounds__(256) void site_kernel(
    const float* __restrict__ phi,      // [NTRIP][8]
    const float* __restrict__ a_ws,     // [4096][64]
    float* __restrict__ out_z,          // [NTRIP][8] complex -> float pairs
    float2* __restrict__ d_site)        // [NTRIP]
{
    const int t = blockIdx.x * 256 + threadIdx.x;     // b-major: t = b*4096+site
    const int site = t & (NSITE - 1);

    float X[8];
    {
        const float4* p = (const float4*)(phi + (size_t)t * 8);
        float4 x0 = p[0], x1 = p[1];
        X[0]=x0.x; X[1]=x0.y; X[2]=x0.z; X[3]=x0.w;
        X[4]=x1.x; X[5]=x1.y; X[6]=x1.z; X[7]=x1.w;
    }
    float A[64];
    {
        const float4* p = (const float4*)(a_ws + (size_t)site * 64);
        #pragma unroll
        for (int q = 0; q < 16; ++q) {
            float4 v = p[q];
            A[4*q+0]=v.x; A[4*q+1]=v.y; A[4*q+2]=v.z; A[4*q+3]=v.w;
        }
    }

    float Y[8], w[8];
    #pragma unroll
    for (int i = 0; i < 8; ++i) {
        float s = 0.0f;
        #pragma unroll
        for (int j = 0; j < 8; ++j) s += A[i*8+j] * X[j];
        Y[i] = s;
    }
    #pragma unroll
    for (int i = 0; i < 8; ++i) {
        float s = 0.0f;
        #pragma unroll
        for (int j = 0; j < 8; ++j) s += A[i*8+j] * Y[j];   // w = a_ (a_ X)
        w[i] = s;
    }
    float yy = 0.0f;
    #pragma unroll
    for (int i = 0; i < 8; ++i) yy += Y[i] * Y[i];
    const float lam = sqrtf(1.0f + yy);
    const float inv = 1.0f / lam;

    // tildeZ = lam*X + i*Y  (interleaved complex64)
    {
        float4* oz = (float4*)(out_z + (size_t)t * 16);
        #pragma unroll
        for (int q = 0; q < 4; ++q)
            oz[q] = make_float4(X[2*q] * lam, Y[2*q], X[2*q+1] * lam, Y[2*q+1]);
    }

    // J and its determinant (complex, fully in registers)
    float2 J[8][8];
    #pragma unroll
    for (int r = 0; r < 8; ++r)
        #pragma unroll
        for (int c = 0; c < 8; ++c)
            J[r][c] = make_float2(((r == c) ? lam : 0.0f) - X[r] * w[c] * inv,
                                  A[r*8 + c]);

    float2 det = make_float2(1.0f, 0.0f);
    #pragma unroll
    for (int k = 0; k < 8; ++k) {
        float2 d = J[k][k];
        det = cmul(det, d);
        float den = 1.0f / (d.x * d.x + d.y * d.y);
        float2 id = make_float2(d.x * den, -d.y * den);
        #pragma unroll
        for (int r = k + 1; r < 8; ++r) {
            float2 f = cmul(J[r][k], id);
            #pragma unroll
            for (int c = k + 1; c < 8; ++c) {
                float2 p = cmul(f, J[k][c]);
                J[r][c].x -= p.x;
                J[r][c].y -= p.y;
            }
        }
    }
    const float s2 = inv * inv;
    d_site[t] = make_float2(det.x * s2, det.y * s2);
}

// ---------------------------------------------------------------------------
// Kernel 3: per-batch complex product over 4096 sites (serial-16 + LDS tree).
// Values are ~1 in magnitude, so the fp32 product is well-conditioned.
// ---------------------------------------------------------------------------
__global__ __launch_bounds__(256) void reduce_kernel(
    const float2* __restrict__ d_site,  // [NB][4096]
    float* __restrict__ out_det)        // [NB] complex -> float pairs
{
    __shared__ float2 sh[256];
    const int b = blockIdx.x;
    const float2* base = d_site + (size_t)b * NSITE;
    float2 p = make_float2(1.0f, 0.0f);
    for (int s = threadIdx.x; s < NSITE; s += 256) p = cmul(p, base[s]);
    sh[threadIdx.x] = p;
    __syncthreads();
    #pragma unroll
    for (int off = 128; off > 0; off >>= 1) {
        if (threadIdx.x < (unsigned)off)
            sh[threadIdx.x] = cmul(sh[threadIdx.x], sh[threadIdx.x + off]);
        __syncthreads();
    }
    if (threadIdx.x == 0) {
        out_det[2*b + 0] = sh[0].x;
        out_det[2*b + 1] = sh[0].y;
    }
}

// ---------------------------------------------------------------------------
// launch: ws layout = [a_ : 4096*64 f32 = 1 MB][d_site : 524288 float2 = 4 MB]
// out   = [tildeZ : 8388608 f32][detJ : 256 f32]
// ---------------------------------------------------------------------------
extern "C" void kernel_launch(void* const* d_in, const int* in_sizes, int n_in,
                              void* d_out, int out_size, void* d_ws, size_t ws_size,
                              hipStream_t stream) {
    (void)in_sizes; (void)n_in; (void)out_size; (void)ws_size;
    const float* phi    = (const float*)d_in[0];   // [128,64,64,8,1] f32
    const float* coeffs = (const float*)d_in[1];   // [64,64,15]      f32
    float*  out    = (float*)d_out;
    float*  a_ws   = (float*)d_ws;
    float2* d_site = (float2*)((char*)d_ws + (size_t)NSITE * 64 * sizeof(float));

    build_a_kernel<<<NSITE / 16 / 4, 128, 0, stream>>>(coeffs, a_ws);
    site_kernel<<<NTRIP / 256, 256, 0, stream>>>(phi, a_ws, out, d_site);
    reduce_kernel<<<NB, 256, 0, stream>>>(d_site, out + (size_t)NTRIP * 16);
}